// PoseGraph_12549894439529
// MI455X (gfx1250) — compile-verified
//
#include <hip/hip_runtime.h>
#include <stdint.h>

#define NFRAMES   64
#define POSE_F    7          // 3 trans + 3 qvec + 1 qw
#define BLOCK     256
#define POSE_BYTES (NFRAMES * POSE_F * 4)   // 1792 bytes = 112 x 16B

__global__ __launch_bounds__(BLOCK) void pose_graph_se3_kernel(
    const float* __restrict__ poses,   // [64,7]
    const float* __restrict__ tc,      // [N,3]
    const float* __restrict__ tw,      // [N,3]
    const int*   __restrict__ idx_lo,  // int64 indices viewed as int32 pairs (LE, low dword valid)
    float*       __restrict__ out,     // [N,3]
    int n_edges)
{
    // Pose table staged in LDS, stride 7 floats. gcd(7,64)==1 so random
    // frame gathers map distinct frames to distinct banks (conflict-free).
    __shared__ float sP[NFRAMES * POSE_F];

    const int tid = threadIdx.x;

    // ---- CDNA5 async global->LDS copy of the pose table ----------------
    // 1792 bytes = 112 lanes x b128. GVS addressing: saddr = pose base,
    // vaddr = 32-bit per-lane byte offset, vdst = LDS byte address.
    if (tid < (POSE_BYTES / 16)) {
        unsigned lds_addr = (unsigned)(size_t)(&sP[0]) + (unsigned)(tid * 16);
        unsigned g_off    = (unsigned)(tid * 16);
        unsigned long long base = (unsigned long long)(size_t)poses;
        asm volatile("global_load_async_to_lds_b128 %0, %1, %2"
                     :
                     : "v"(lds_addr), "v"(g_off), "s"(base)
                     : "memory");
    }
    asm volatile("s_wait_asynccnt 0" ::: "memory");
    __syncthreads();
    // --------------------------------------------------------------------

    const int e = blockIdx.x * BLOCK + tid;
    if (e >= n_edges) return;

    // Index values are in [0,64): only the low dword of the int64 matters.
    const int f = __builtin_nontemporal_load(&idx_lo[2 * e]);

    const float* P = &sP[f * POSE_F];
    const float ttx = P[0], tty = P[1], ttz = P[2];
    const float qx  = P[3], qy  = P[4], qz  = P[5], qw = P[6];

    // Streaming (no-reuse) loads -> non-temporal, coalesced 12B/lane.
    const float* pc = tc + 3 * (size_t)e;
    const float px = __builtin_nontemporal_load(pc + 0);
    const float py = __builtin_nontemporal_load(pc + 1);
    const float pz = __builtin_nontemporal_load(pc + 2);

    // uv = qv x p
    const float ux = qy * pz - qz * py;
    const float uy = qz * px - qx * pz;
    const float uz = qx * py - qy * px;
    // w = uv + qw * p
    const float wx = fmaf(qw, px, ux);
    const float wy = fmaf(qw, py, uy);
    const float wz = fmaf(qw, pz, uz);
    // r = p + 2 * (qv x w)
    const float rx = fmaf(2.0f, qy * wz - qz * wy, px);
    const float ry = fmaf(2.0f, qz * wx - qx * wz, py);
    const float rz = fmaf(2.0f, qx * wy - qy * wx, pz);

    const float* pw = tw + 3 * (size_t)e;
    const float ox = (rx + ttx) - __builtin_nontemporal_load(pw + 0);
    const float oy = (ry + tty) - __builtin_nontemporal_load(pw + 1);
    const float oz = (rz + ttz) - __builtin_nontemporal_load(pw + 2);

    float* po = out + 3 * (size_t)e;
    __builtin_nontemporal_store(ox, po + 0);
    __builtin_nontemporal_store(oy, po + 1);
    __builtin_nontemporal_store(oz, po + 2);
}

extern "C" void kernel_launch(void* const* d_in, const int* in_sizes, int n_in,
                              void* d_out, int out_size, void* d_ws, size_t ws_size,
                              hipStream_t stream) {
    const float* poses = (const float*)d_in[0];   // [64,7] f32
    const float* tc    = (const float*)d_in[1];   // [N,3]  f32
    const float* tw    = (const float*)d_in[2];   // [N,3]  f32
    const int*   idx   = (const int*)d_in[3];     // [N]    int64 (read low dwords)
    float*       out   = (float*)d_out;           // [N,3]  f32

    const int n = in_sizes[3];                    // N_EDGES
    const int blocks = (n + BLOCK - 1) / BLOCK;

    hipLaunchKernelGGL(pose_graph_se3_kernel, dim3(blocks), dim3(BLOCK), 0, stream,
                       poses, tc, tw, idx, out, n);
}